// Median_filter_7705171329347
// MI455X (gfx1250) — compile-verified
//
#include <hip/hip_runtime.h>
#include <hip/hip_bf16.h>

// Problem constants (16,3,512,512) f32
#define IMG_W   512
#define IMG_H   512
#define PLANES  48            // 16 * 3
#define BDIM    128           // 4 wave32 per block; each thread -> 4 pixels (float4)

#ifndef HAVE_ASYNC_LDS
#  if defined(__has_builtin)
#    if __has_builtin(__builtin_amdgcn_global_load_async_to_lds_b128)
#      define HAVE_ASYNC_LDS 1
#    else
#      define HAVE_ASYNC_LDS 0
#    endif
#  else
#    define HAVE_ASYNC_LDS 0
#  endif
#endif

// v4i32 pointers in explicit address spaces, matching the builtin's signature:
//   __builtin_amdgcn_global_load_async_to_lds_b128(v4i AS1* src, v4i AS3* dst, imm off, imm cpol)
typedef int v4i __attribute__((ext_vector_type(4)));
typedef __attribute__((address_space(1))) v4i gv4i;   // global
typedef __attribute__((address_space(3))) v4i lv4i;   // LDS

__device__ __forceinline__ void cswap(float& a, float& b) {
    float t = fminf(a, b);
    b = fmaxf(a, b);
    a = t;
}

// Exact median of 9 via 19-compare-exchange sorting network.
__device__ __forceinline__ float median9(float p0, float p1, float p2,
                                         float p3, float p4, float p5,
                                         float p6, float p7, float p8) {
    cswap(p1, p2); cswap(p4, p5); cswap(p7, p8);
    cswap(p0, p1); cswap(p3, p4); cswap(p6, p7);
    cswap(p1, p2); cswap(p4, p5); cswap(p7, p8);
    cswap(p0, p3); cswap(p5, p8); cswap(p4, p7);
    cswap(p3, p6); cswap(p1, p4); cswap(p2, p5);
    cswap(p4, p7); cswap(p4, p2); cswap(p6, p4);
    cswap(p4, p2);
    return p4;
}

__global__ __launch_bounds__(BDIM)
void median3x3_kernel(const float* __restrict__ src, float* __restrict__ dst) {
    __shared__ __align__(16) float rows[3][IMG_W];

    const int t  = threadIdx.x;        // 0..127
    const int h  = blockIdx.x;         // 0..511  (output row)
    const int pl = blockIdx.y;         // 0..47   (plane = b*C + c)
    const int x0 = t * 4;              // first of 4 output columns

    const size_t planeBase = (size_t)pl * ((size_t)IMG_H * IMG_W);

    // Stage rows h-1, h, h+1 into LDS (zero padding for out-of-image rows).
#pragma unroll
    for (int i = 0; i < 3; ++i) {
        const int hr = h - 1 + i;
        if (hr >= 0 && hr < IMG_H) {
            const float* g = src + planeBase + (size_t)hr * IMG_W + x0;
#if HAVE_ASYNC_LDS
            // CDNA5 async copy: global -> LDS without a VGPR round trip (ASYNCcnt).
            __builtin_amdgcn_global_load_async_to_lds_b128(
                (gv4i*)g, (lv4i*)&rows[i][x0], 0, 0);
#else
            *(float4*)&rows[i][x0] = *(const float4*)g;
#endif
        } else {
            float4 z = {0.f, 0.f, 0.f, 0.f};
            *(float4*)&rows[i][x0] = z;
        }
    }
#if HAVE_ASYNC_LDS
    asm volatile("s_wait_asynccnt 0" ::: "memory");
#endif
    __syncthreads();

    // Pull a 6-wide window per row: columns x0-1 .. x0+4 (zero at image edges).
    float v[3][6];
#pragma unroll
    for (int i = 0; i < 3; ++i) {
        v[i][0] = (x0 == 0) ? 0.f : rows[i][x0 - 1];
        float4 m = *(const float4*)&rows[i][x0];
        v[i][1] = m.x; v[i][2] = m.y; v[i][3] = m.z; v[i][4] = m.w;
        v[i][5] = (x0 + 4 >= IMG_W) ? 0.f : rows[i][x0 + 4];
    }

    float r[4];
#pragma unroll
    for (int j = 0; j < 4; ++j) {
        r[j] = median9(v[0][j], v[0][j + 1], v[0][j + 2],
                       v[1][j], v[1][j + 1], v[1][j + 2],
                       v[2][j], v[2][j + 1], v[2][j + 2]);
    }

    float4 out;
    out.x = r[0]; out.y = r[1]; out.z = r[2]; out.w = r[3];
    *(float4*)(dst + planeBase + (size_t)h * IMG_W + x0) = out;
}

__global__ __launch_bounds__(256)
void copy4_kernel(const float4* __restrict__ src, float4* __restrict__ dst, int n4) {
    int i = blockIdx.x * blockDim.x + threadIdx.x;
    if (i < n4) dst[i] = src[i];
}

extern "C" void kernel_launch(void* const* d_in, const int* in_sizes, int n_in,
                              void* d_out, int out_size, void* d_ws, size_t ws_size,
                              hipStream_t stream) {
    const float* noised = (const float*)d_in[0];
    const float* cover  = (const float*)d_in[1];
    float* out = (float*)d_out;

    const int N = PLANES * IMG_H * IMG_W;   // 12,582,912

    // Median blur: one block per (row, plane).
    dim3 grid(IMG_H, PLANES);
    median3x3_kernel<<<grid, BDIM, 0, stream>>>(noised, out);

    // Pass-through cover into second half of the output.
    const int n4 = N / 4;
    copy4_kernel<<<(n4 + 255) / 256, 256, 0, stream>>>(
        (const float4*)cover, (float4*)(out + N), n4);
}